// My_InteractionNetwork_59004260712594
// MI455X (gfx1250) — compile-verified
//
#include <hip/hip_runtime.h>
#include <hip/hip_bf16.h>

#define NNODES 100000
#define NEDGES 1600000
#define HID    100
#define NT     7      // 7 n-tiles of 16 -> 112 padded columns
#define TS     116    // LDS tile row stride (floats), breaks bank pattern

typedef float v2f __attribute__((ext_vector_type(2)));
typedef float v8f __attribute__((ext_vector_type(8)));

// ---------------------------------------------------------------------------
// Zero the aggregation buffer (must be re-zeroed every deterministic call).
// ---------------------------------------------------------------------------
__global__ void zero_kernel(float* __restrict__ p, size_t n) {
    size_t i = (size_t)blockIdx.x * blockDim.x + threadIdx.x;
    size_t stride = (size_t)gridDim.x * blockDim.x;
    for (; i < n; i += stride) p[i] = 0.0f;
}

// ---------------------------------------------------------------------------
// Pack a row-major weight matrix W[K][NC] into WMMA B-fragment order:
// tile t = (kstep*NT + ntile), lane l (h=l>>4, n=l&15):
//   dst[2t+0] = W[kstep*4 + 2h    ][ntile*16 + n]
//   dst[2t+1] = W[kstep*4 + 2h + 1][ntile*16 + n]
// Zero-filled outside K x NC (K padding and N padding to 112).
// Matches V_WMMA B layout: VGPR0 = rows {k0,k0+2} per lane half, VGPR1 = {k0+1,k0+3}.
// ---------------------------------------------------------------------------
__global__ void pack_w(const float* __restrict__ src, int K, int NC, int ksteps,
                       float* __restrict__ dst) {
    int t = blockIdx.x * blockDim.x + threadIdx.x;
    int total = ksteps * NT * 32;
    if (t >= total) return;
    int kstep = t / (NT * 32);
    int rem   = t % (NT * 32);
    int nt    = rem >> 5;
    int lane  = rem & 31;
    int h = lane >> 4, n = lane & 15;
    int k0  = kstep * 4 + 2 * h;
    int col = nt * 16 + n;
    float v0 = (k0     < K && col < NC) ? src[(size_t)k0 * NC + col]       : 0.0f;
    float v1 = (k0 + 1 < K && col < NC) ? src[(size_t)(k0 + 1) * NC + col] : 0.0f;
    dst[2 * t + 0] = v0;
    dst[2 * t + 1] = v1;
}

// ---------------------------------------------------------------------------
// One dense layer on a per-wave 16-row tile held in LDS:
//   T[16, K] (K = ksteps*4, zero-padded) x Wp[K, 112]  -> relu(+bias) -> T[16, 112]
// A operand : 8B LDS load per lane  (A layout: lane = M, halves split K)
// B operand : 8B coalesced global load from pre-packed fragments
// D -> tile : C/D layout (VGPR j: M = 8*h + j, N = lane&15) via LDS stores
// ---------------------------------------------------------------------------
__device__ __forceinline__ void mlp_layer(float* __restrict__ T,
                                          const float* __restrict__ wp,
                                          const float* __restrict__ bias,
                                          int biasN, int ksteps, int lane) {
    const int m = lane & 15;
    const int h = lane >> 4;
    v8f acc[NT];
#pragma unroll
    for (int i = 0; i < NT; ++i) acc[i] = (v8f){0.f, 0.f, 0.f, 0.f, 0.f, 0.f, 0.f, 0.f};

    const v2f* __restrict__ wp2 = (const v2f*)wp;
    for (int ks = 0; ks < ksteps; ++ks) {
        v2f a = *(const v2f*)(T + m * TS + ks * 4 + 2 * h);  // 8B aligned
#pragma unroll
        for (int nt = 0; nt < NT; ++nt) {
            v2f b = wp2[(ks * NT + nt) * 32 + lane];         // coalesced b64
            acc[nt] = __builtin_amdgcn_wmma_f32_16x16x4_f32(
                /*neg_a=*/false, a, /*neg_b=*/false, b,
                /*c_mod=*/(short)0, acc[nt],
                /*reuse_a=*/false, /*reuse_b=*/false);
        }
    }
    __syncthreads();  // all tile reads done before overwrite
#pragma unroll
    for (int nt = 0; nt < NT; ++nt) {
        int col = nt * 16 + m;
        float bv = (col < biasN) ? bias[col] : 0.0f;
#pragma unroll
        for (int j = 0; j < 8; ++j) {
            float v = acc[nt][j] + bv;
            T[(h * 8 + j) * TS + col] = v > 0.0f ? v : 0.0f;  // ReLU (all fused layers use it)
        }
    }
    __syncthreads();
}

// ---------------------------------------------------------------------------
// Edge kernel: 16 edges per wave, 8 waves per block.
// Stages [x_s, x_r, 0] (K=5 padded to 8), runs 4 fused WMMA layers, then
// scatter-adds the 16x100 effects onto agg[sender] with f32 atomics.
// ---------------------------------------------------------------------------
__global__ void __launch_bounds__(256) edge_kernel(
    const float* __restrict__ x, const int* __restrict__ ei,
    const float* __restrict__ w1p, const float* __restrict__ b1,
    const float* __restrict__ w2p, const float* __restrict__ b2,
    const float* __restrict__ w3p, const float* __restrict__ b3,
    const float* __restrict__ w4p, const float* __restrict__ b4,
    float* __restrict__ agg) {
    __shared__ float tiles[8 * 16 * TS];
    const int lane = threadIdx.x & 31;
    const int wave = threadIdx.x >> 5;
    float* T = tiles + wave * 16 * TS;
    const int e0 = (blockIdx.x * 8 + wave) * 16;
    const int m = lane & 15;

    const float2* __restrict__ x2 = (const float2*)x;
    if (lane < 16) {
        int s = ei[e0 + m];                 // sender row
        float2 xs = x2[s];
        T[m * TS + 0] = xs.x; T[m * TS + 1] = xs.y;
        T[m * TS + 4] = 0.f;                // r_info = 0
        T[m * TS + 5] = 0.f; T[m * TS + 6] = 0.f; T[m * TS + 7] = 0.f; // K pad
    } else {
        int r = ei[NEDGES + e0 + m];        // receiver row
        float2 xr = x2[r];
        T[m * TS + 2] = xr.x; T[m * TS + 3] = xr.y;
    }
    __syncthreads();

    mlp_layer(T, w1p, b1, HID, 2,  lane);   // 5(->8) -> 100
    mlp_layer(T, w2p, b2, HID, 25, lane);   // 100 -> 100
    mlp_layer(T, w3p, b3, HID, 25, lane);   // 100 -> 100
    mlp_layer(T, w4p, b4, HID, 25, lane);   // 100 -> 100 (effects, relu'd)

    // scatter-add onto SENDER nodes (reference quirk)
#pragma unroll 4
    for (int et = 0; et < 16; ++et) {
        int s = ei[e0 + et];
        size_t base = (size_t)s * HID;
        for (int c = lane; c < HID; c += 32)
            atomicAdd(&agg[base + c], T[et * TS + c]);
    }
}

// ---------------------------------------------------------------------------
// Node kernel: 16 nodes per wave. Stage [x(2) | agg(100) | 0 0] (K=102->104),
// WMMA layer 102->100 + ReLU, then VALU 100->2 (one (node,dim) per lane).
// ---------------------------------------------------------------------------
__global__ void __launch_bounds__(256) node_kernel(
    const float* __restrict__ x, const float* __restrict__ agg,
    const float* __restrict__ ow1p, const float* __restrict__ ob1,
    const float* __restrict__ ow2, const float* __restrict__ ob2,
    float* __restrict__ out) {
    __shared__ float tiles[8 * 16 * TS];
    const int lane = threadIdx.x & 31;
    const int wave = threadIdx.x >> 5;
    float* T = tiles + wave * 16 * TS;
    const int n0 = (blockIdx.x * 8 + wave) * 16;
    const bool active = (n0 < NNODES);      // wave-uniform

    if (active) {
        for (int rr = 0; rr < 16; ++rr) {
            size_t base = (size_t)(n0 + rr) * HID;
            for (int c = lane; c < HID; c += 32)     // coalesced agg rows
                T[rr * TS + 2 + c] = agg[base + c];
        }
        if (lane < 16) {
            int node = n0 + lane;
            T[lane * TS + 0]   = x[node * 2 + 0];
            T[lane * TS + 1]   = x[node * 2 + 1];
            T[lane * TS + 102] = 0.f;                // K pad to 104
            T[lane * TS + 103] = 0.f;
        }
    }
    __syncthreads();
    // All waves run the layer (barriers must be uniform); inactive waves
    // crunch their own garbage tile and never write global memory.
    mlp_layer(T, ow1p, ob1, HID, 26, lane);          // 102(->104) -> 100, ReLU

    if (active) {
        int m2 = lane >> 1, dim = lane & 1;          // 16 nodes x 2 dims = 32 lanes
        float sum = ob2[dim];
        for (int k = 0; k < HID; ++k)
            sum += T[m2 * TS + k] * ow2[k * 2 + dim];
        out[(size_t)(n0 + m2) * 2 + dim] = sum;      // no ReLU on final layer
    }
}

// ---------------------------------------------------------------------------
extern "C" void kernel_launch(void* const* d_in, const int* in_sizes, int n_in,
                              void* d_out, int out_size, void* d_ws, size_t ws_size,
                              hipStream_t stream) {
    const float* x     = (const float*)d_in[0];
    const int*   ei    = (const int*)  d_in[1];
    const float* rm_w1 = (const float*)d_in[2];
    const float* rm_b1 = (const float*)d_in[3];
    const float* rm_w2 = (const float*)d_in[4];
    const float* rm_b2 = (const float*)d_in[5];
    const float* rm_w3 = (const float*)d_in[6];
    const float* rm_b3 = (const float*)d_in[7];
    const float* rm_w4 = (const float*)d_in[8];
    const float* rm_b4 = (const float*)d_in[9];
    const float* om_w1 = (const float*)d_in[10];
    const float* om_b1 = (const float*)d_in[11];
    const float* om_w2 = (const float*)d_in[12];
    const float* om_b2 = (const float*)d_in[13];

    float* ws   = (float*)d_ws;
    float* agg  = ws;                                   // 100000*100 = 10,000,000 floats
    float* W1P  = ws + (size_t)NNODES * HID;            // 2*7*64  = 896
    float* W2P  = W1P + 2  * NT * 64;                   // 25*7*64 = 11200
    float* W3P  = W2P + 25 * NT * 64;
    float* W4P  = W3P + 25 * NT * 64;
    float* OW1P = W4P + 25 * NT * 64;                   // 26*7*64 = 11648

    zero_kernel<<<2048, 256, 0, stream>>>(agg, (size_t)NNODES * HID);

    pack_w<<<(2  * NT * 32 + 255) / 256, 256, 0, stream>>>(rm_w1, 5,   HID, 2,  W1P);
    pack_w<<<(25 * NT * 32 + 255) / 256, 256, 0, stream>>>(rm_w2, 100, HID, 25, W2P);
    pack_w<<<(25 * NT * 32 + 255) / 256, 256, 0, stream>>>(rm_w3, 100, HID, 25, W3P);
    pack_w<<<(25 * NT * 32 + 255) / 256, 256, 0, stream>>>(rm_w4, 100, HID, 25, W4P);
    pack_w<<<(26 * NT * 32 + 255) / 256, 256, 0, stream>>>(om_w1, 102, HID, 26, OW1P);

    // 1.6M edges / (8 waves * 16 edges) = 12500 blocks exactly
    edge_kernel<<<12500, 256, 0, stream>>>(x, ei, W1P, rm_b1, W2P, rm_b2,
                                           W3P, rm_b3, W4P, rm_b4, agg);
    // 100000 nodes / 16 = 6250 wave-tiles -> 782 blocks of 8 waves
    node_kernel<<<(6250 + 7) / 8, 256, 0, stream>>>(x, agg, OW1P, om_b1,
                                                    om_w2, om_b2, (float*)d_out);
}